// AttnDecoderRNN_86887188398769
// MI455X (gfx1250) — compile-verified
//
#include <hip/hip_runtime.h>
#include <math.h>

#define VV 50257
#define EE 1024
#define HH 2048
#define VISZ 2048
#define NBX 36
#define LLX 50

typedef __attribute__((ext_vector_type(2))) float v2f;
typedef __attribute__((ext_vector_type(8))) float v8f;

// ---------------- wave / block reductions (wave32) ----------------
__device__ __forceinline__ float wave_allsum(float v) {
#pragma unroll
  for (int m = 16; m > 0; m >>= 1) v += __shfl_xor(v, m, 32);
  return v;
}
__device__ __forceinline__ float wave_allmax(float v) {
#pragma unroll
  for (int m = 16; m > 0; m >>= 1) v = fmaxf(v, __shfl_xor(v, m, 32));
  return v;
}

// block-stride partial dot over n floats (n % 4 == 0), float4 loads
__device__ __forceinline__ float blockdot_part(const float* __restrict__ w,
                                               const float* __restrict__ a, int n) {
  const float4* w4 = (const float4*)w;
  const float4* a4 = (const float4*)a;
  float acc = 0.f;
  for (int i = threadIdx.x; i < (n >> 2); i += blockDim.x) {
    float4 x = w4[i], y = a4[i];
    acc = fmaf(x.x, y.x, acc); acc = fmaf(x.y, y.y, acc);
    acc = fmaf(x.z, y.z, acc); acc = fmaf(x.w, y.w, acc);
  }
  return acc;
}
__device__ __forceinline__ float block_reduce_sum(float v) {
  __shared__ float s[8];
  int lane = threadIdx.x & 31, wv = threadIdx.x >> 5;
  v = wave_allsum(v);
  if (lane == 0) s[wv] = v;
  __syncthreads();
  float t = (lane < (int)(blockDim.x >> 5)) ? s[lane] : 0.f;
  return wave_allsum(t);
}
// wave-stride (lane) partial dot, float4 loads
__device__ __forceinline__ float wavedot_part(const float* __restrict__ w,
                                              const float* __restrict__ a, int n, int lane) {
  const float4* w4 = (const float4*)w;
  const float4* a4 = (const float4*)a;
  float acc = 0.f;
  for (int i = lane; i < (n >> 2); i += 32) {
    float4 x = w4[i], y = a4[i];
    acc = fmaf(x.x, y.x, acc); acc = fmaf(x.y, y.y, acc);
    acc = fmaf(x.z, y.z, acc); acc = fmaf(x.w, y.w, acc);
  }
  return acc;
}

// ---------------- 1. embedding gather ----------------
__global__ void k_embed(const float* __restrict__ w2v, const int* __restrict__ ids,
                        float* __restrict__ emb) {
  int i = blockIdx.x * blockDim.x + threadIdx.x;
  if (i < EE) emb[i] = w2v[(size_t)ids[0] * EE + i];
}

// ---------------- 2. attention logits: row per block ----------------
__global__ void k_attn_logits(const float* __restrict__ W, const float* __restrict__ b,
                              const float* __restrict__ emb, const float* __restrict__ h0,
                              float* __restrict__ out) {
  int row = blockIdx.x;
  const float* wr = W + (size_t)row * (EE + HH);
  float acc = blockdot_part(wr, emb, EE) + blockdot_part(wr + EE, h0, HH);
  acc = block_reduce_sum(acc);
  if (threadIdx.x == 0) out[row] = acc + b[row];
}

// ---------------- 3. visual attention logits ----------------
__global__ void k_vis_logits(const float* __restrict__ W, const float* __restrict__ b,
                             const float* __restrict__ h0, const float* __restrict__ visf,
                             float* __restrict__ out) {
  int row = blockIdx.x;
  const float* wr = W + (size_t)row * (HH + NBX * VISZ);
  float acc = blockdot_part(wr, h0, HH) + blockdot_part(wr + HH, visf, NBX * VISZ);
  acc = block_reduce_sum(acc);
  if (threadIdx.x == 0) out[row] = acc + b[row];
}

// ---------------- 4. two tiny softmaxes (wave0: text, wave1: visual) ----------------
__global__ void k_softmax(const float* __restrict__ alog, const float* __restrict__ vlog,
                          float* __restrict__ aw, float* __restrict__ vw,
                          float* __restrict__ out_aw) {
  int wv = threadIdx.x >> 5, lane = threadIdx.x & 31;
  if (wv == 0) {
    float v0 = (lane < LLX) ? alog[lane] : -INFINITY;
    float v1 = (lane + 32 < LLX) ? alog[lane + 32] : -INFINITY;
    float m = wave_allmax(fmaxf(v0, v1));
    float e0 = (lane < LLX) ? expf(v0 - m) : 0.f;
    float e1 = (lane + 32 < LLX) ? expf(v1 - m) : 0.f;
    float s = wave_allsum(e0 + e1);
    if (lane < LLX)      { float r = e0 / s; aw[lane] = r;      out_aw[lane] = r; }
    if (lane + 32 < LLX) { float r = e1 / s; aw[lane + 32] = r; out_aw[lane + 32] = r; }
  } else if (wv == 1) {
    float v0 = (lane < NBX) ? vlog[lane] : -INFINITY;
    float v1 = (lane + 32 < NBX) ? vlog[lane + 32] : -INFINITY;
    float m = wave_allmax(fmaxf(v0, v1));
    float e0 = (lane < NBX) ? expf(v0 - m) : 0.f;
    float e1 = (lane + 32 < NBX) ? expf(v1 - m) : 0.f;
    float s = wave_allsum(e0 + e1);
    if (lane < NBX)      vw[lane] = e0 / s;
    if (lane + 32 < NBX) vw[lane + 32] = e1 / s;
  }
}

// ---------------- 5. contexts: attn_applied[1024] + fused avgpool vis_pooled[1024] ----------------
__global__ void k_context(const float* __restrict__ aw, const float* __restrict__ enc,
                          const float* __restrict__ vw, const float* __restrict__ visf,
                          float* __restrict__ attn_applied, float* __restrict__ vis_pooled) {
  int j = blockIdx.x * blockDim.x + threadIdx.x;  // 0..1023
  if (j >= HH / 2) return;
  float s = 0.f;
  for (int l = 0; l < LLX; ++l) s = fmaf(aw[l], enc[l * (HH / 2) + j], s);
  attn_applied[j] = s;
  float p = 0.f;
  for (int bx = 0; bx < NBX; ++bx) {
    const float2* v2 = (const float2*)(visf + (size_t)bx * VISZ);
    float2 t = v2[j];
    p = fmaf(vw[bx], 0.5f * (t.x + t.y), p);
  }
  vis_pooled[j] = p;
}

// ---------------- 6. generic relu(W @ [a0;a1] + b), wave per row ----------------
__global__ void k_gemv_relu2(const float* __restrict__ W, const float* __restrict__ b,
                             const float* __restrict__ a0, int n0,
                             const float* __restrict__ a1, int n1,
                             float* __restrict__ out, int rows) {
  int wv = blockIdx.x * (blockDim.x >> 5) + (threadIdx.x >> 5);
  int lane = threadIdx.x & 31;
  if (wv >= rows) return;
  const float* wr = W + (size_t)wv * (n0 + n1);
  float acc = wavedot_part(wr, a0, n0, lane) + wavedot_part(wr + n0, a1, n1, lane);
  acc = wave_allsum(acc);
  if (lane == 0) out[wv] = fmaxf(acc + b[wv], 0.f);
}

// ---------------- 7. fused GRU cell, wave per hidden element ----------------
__global__ void k_gru(const float* __restrict__ Wih, const float* __restrict__ Whh,
                      const float* __restrict__ bih, const float* __restrict__ bhh,
                      const float* __restrict__ x, const float* __restrict__ hid,
                      float* __restrict__ hnew, float* __restrict__ out_h) {
  int j = blockIdx.x * (blockDim.x >> 5) + (threadIdx.x >> 5);
  int lane = threadIdx.x & 31;
  if (j >= HH) return;
  float gir = wave_allsum(wavedot_part(Wih + (size_t)j * EE, x, EE, lane)) + bih[j];
  float giz = wave_allsum(wavedot_part(Wih + (size_t)(j + HH) * EE, x, EE, lane)) + bih[j + HH];
  float gin = wave_allsum(wavedot_part(Wih + (size_t)(j + 2 * HH) * EE, x, EE, lane)) + bih[j + 2 * HH];
  float ghr = wave_allsum(wavedot_part(Whh + (size_t)j * HH, hid, HH, lane)) + bhh[j];
  float ghz = wave_allsum(wavedot_part(Whh + (size_t)(j + HH) * HH, hid, HH, lane)) + bhh[j + HH];
  float ghn = wave_allsum(wavedot_part(Whh + (size_t)(j + 2 * HH) * HH, hid, HH, lane)) + bhh[j + 2 * HH];
  float r = 1.f / (1.f + expf(-(gir + ghr)));
  float z = 1.f / (1.f + expf(-(giz + ghz)));
  float n = tanhf(fmaf(r, ghn, gin));
  float h = fmaf(z, hid[j], (1.f - z) * n);
  if (lane == 0) { hnew[j] = h; out_h[j] = h; }
}

// ---------------- 8. big GEMV logits = out_W @ hnew + b via V_WMMA_F32_16X16X4_F32 ----------------
// Each wave owns a 16-row tile of out_W. A = 16x4 f32 tile (ISA layout: lanes 0-15 carry M,
// half-wave selects K pair). B has hnew in column N=0 only (lane 0 -> K0,K1; lane 16 -> K2,K3).
// After the K sweep, D column 0 holds the 16 dot products: lane 0 c[0..7]=M0..7, lane 16 c[0..7]=M8..15.
__global__ void k_logits_wmma(const float* __restrict__ W, const float* __restrict__ b,
                              const float* __restrict__ h, float* __restrict__ logits) {
  int lane = threadIdx.x & 31;
  int wv = threadIdx.x >> 5;
  int tile = blockIdx.x * 8 + wv;
  int m0 = tile * 16;
  int row = m0 + (lane & 15);
  int rowc = row < VV ? row : VV - 1;         // clamp; invalid rows land in unstored D slots
  int koff = (lane >> 4) << 1;                // 0 for lanes 0-15, 2 for lanes 16-31
  const float* wp = W + (size_t)rowc * HH + koff;
  const float* hp = h + koff;
#if __has_builtin(__builtin_amdgcn_wmma_f32_16x16x4_f32)
  bool isHead = (lane & 15) == 0;
  v8f c = {0.f, 0.f, 0.f, 0.f, 0.f, 0.f, 0.f, 0.f};
  for (int k0 = 0; k0 < HH; k0 += 4) {
    float2 av = *(const float2*)wp; wp += 4;
    float2 hv = *(const float2*)hp; hp += 4;
    v2f a;  a.x = av.x;  a.y = av.y;
    v2f bb; bb.x = isHead ? hv.x : 0.f; bb.y = isHead ? hv.y : 0.f;
    c = __builtin_amdgcn_wmma_f32_16x16x4_f32(false, a, false, bb, (short)0, c,
                                              false, false);
  }
  if ((lane & 15) == 0) {
    int base = m0 + (lane >> 4) * 8;
#pragma unroll
    for (int r = 0; r < 8; ++r) {
      int idx = base + r;
      if (idx < VV) logits[idx] = c[r] + b[idx];
    }
  }
#else
  // scalar fallback: lane l and l+16 split K of row m0+(l&15)
  float acc = 0.f;
  for (int k0 = 0; k0 < HH; k0 += 4) {
    float2 av = *(const float2*)wp; wp += 4;
    float2 hv = *(const float2*)hp; hp += 4;
    acc = fmaf(av.x, hv.x, acc); acc = fmaf(av.y, hv.y, acc);
  }
  acc += __shfl_xor(acc, 16, 32);
  if (lane < 16 && row < VV) logits[row] = acc + b[row];
#endif
}

// ---------------- 9. log-sum-exp reduce over V ----------------
__global__ void k_lse(const float* __restrict__ logits, float* __restrict__ red) {
  __shared__ float s[32];
  int lane = threadIdx.x & 31, wv = threadIdx.x >> 5;
  float m = -INFINITY;
  for (int i = threadIdx.x; i < VV; i += 1024) m = fmaxf(m, logits[i]);
  m = wave_allmax(m);
  if (lane == 0) s[wv] = m;
  __syncthreads();
  float M = wave_allmax(s[lane]);
  __syncthreads();
  float sum = 0.f;
  for (int i = threadIdx.x; i < VV; i += 1024) sum += expf(logits[i] - M);
  sum = wave_allsum(sum);
  if (lane == 0) s[wv] = sum;
  __syncthreads();
  float S = wave_allsum(s[lane]);
  if (threadIdx.x == 0) { red[0] = M; red[1] = logf(S); }
}

// ---------------- 10. finalize log-softmax ----------------
__global__ void k_final(const float* __restrict__ logits, const float* __restrict__ red,
                        float* __restrict__ out) {
  int i = blockIdx.x * blockDim.x + threadIdx.x;
  if (i < VV) out[i] = logits[i] - red[0] - red[1];
}

extern "C" void kernel_launch(void* const* d_in, const int* in_sizes, int n_in,
                              void* d_out, int out_size, void* d_ws, size_t ws_size,
                              hipStream_t stream) {
  const int*   ids        = (const int*)  d_in[0];
  const float* h0         = (const float*)d_in[1];   // hidden (1,1,H)
  const float* enc        = (const float*)d_in[2];   // (L, H/2)
  const float* visf       = (const float*)d_in[3];   // (NB, VIS)
  const float* w2v        = (const float*)d_in[4];   // (V, E)
  const float* attn_W     = (const float*)d_in[5];
  const float* attn_b     = (const float*)d_in[6];
  const float* vis_attn_W = (const float*)d_in[7];
  const float* vis_attn_b = (const float*)d_in[8];
  const float* vis_comb_W = (const float*)d_in[9];
  const float* vis_comb_b = (const float*)d_in[10];
  const float* attn_comb_W= (const float*)d_in[11];
  const float* attn_comb_b= (const float*)d_in[12];
  const float* gru_Wih    = (const float*)d_in[13];
  const float* gru_Whh    = (const float*)d_in[14];
  const float* gru_bih    = (const float*)d_in[15];
  const float* gru_bhh    = (const float*)d_in[16];
  const float* out_W      = (const float*)d_in[17];
  const float* out_b      = (const float*)d_in[18];
  float* out = (float*)d_out;   // [ logprobs(V) | hnew(H) | attn_weights(L) ]

  float* ws           = (float*)d_ws;
  float* emb          = ws;            // 1024
  float* alog         = ws + 1024;     // 64
  float* vlog         = ws + 1088;     // 64
  float* aw           = ws + 1152;     // 64
  float* vw           = ws + 1216;     // 64
  float* attn_applied = ws + 1280;     // 1024
  float* vis_pooled   = ws + 2304;     // 1024
  float* hid          = ws + 3328;     // 2048
  float* x            = ws + 5376;     // 1024
  float* hnew         = ws + 6400;     // 2048
  float* red          = ws + 8448;     // 2
  float* logits       = ws + 8704;     // 50257

  k_embed<<<(EE + 255) / 256, 256, 0, stream>>>(w2v, ids, emb);
  k_attn_logits<<<LLX, 256, 0, stream>>>(attn_W, attn_b, emb, h0, alog);
  k_vis_logits<<<NBX, 256, 0, stream>>>(vis_attn_W, vis_attn_b, h0, visf, vlog);
  k_softmax<<<1, 64, 0, stream>>>(alog, vlog, aw, vw, out + VV + HH);
  k_context<<<(HH / 2 + 255) / 256, 256, 0, stream>>>(aw, enc, vw, visf, attn_applied, vis_pooled);
  k_gemv_relu2<<<HH / 8, 256, 0, stream>>>(vis_comb_W, vis_comb_b, h0, HH, vis_pooled, HH / 2, hid, HH);
  k_gemv_relu2<<<EE / 8, 256, 0, stream>>>(attn_comb_W, attn_comb_b, emb, EE, attn_applied, HH / 2, x, EE);
  k_gru<<<HH / 8, 256, 0, stream>>>(gru_Wih, gru_Whh, gru_bih, gru_bhh, x, hid, hnew, out + VV);
  int tiles = (VV + 15) / 16;                 // 3142
  k_logits_wmma<<<(tiles + 7) / 8, 256, 0, stream>>>(out_W, out_b, hnew, logits);
  k_lse<<<1, 1024, 0, stream>>>(logits, red);
  k_final<<<(VV + 255) / 256, 256, 0, stream>>>(logits, red, out);
}